// HGNNModel_4355096839063
// MI455X (gfx1250) — compile-verified
//
#include <hip/hip_runtime.h>

// HGCN propagation on MI455X (gfx1250, wave32).
//   layer k<2 : e = leaky_relu( A @ (A^T @ e), 0.5 )
//   layer 2   : out = LayerNorm( A @ (A^T @ e) ) + ego
// SpMM = gather (global_load_b128) + scatter (global_atomic_add_f32), L2-resident.
// LayerNorm row reductions use V_WMMA_F32_16X16X4_F32 with a ones B-matrix.

#define N_NODES 150000
#define DIM 64
#define LN_EPS 1e-5f
#define LEAKY_SLOPE 0.5f

typedef float v2f __attribute__((ext_vector_type(2)));
typedef float v8f __attribute__((ext_vector_type(8)));

// ---------------------------------------------------------------------------
// Scatter SpMM:  y[dst[e], :] += vals[e] * x[src[e], :]   (D = 64)
// 16 lanes per edge, one float4 per lane: gather is one global_load_b128,
// scatter is 4 non-returning global_atomic_add_f32. x (38.4 MB) and the edge
// lists (57.6 MB) both fit in the 192 MB L2, so this runs at L2 rates.
// ---------------------------------------------------------------------------
__global__ __launch_bounds__(256) void spmm_scatter_kernel(
    const int* __restrict__ dst, const int* __restrict__ src,
    const float* __restrict__ vals, const float* __restrict__ x,
    float* __restrict__ y, int nE)
{
  const long long gid = (long long)blockIdx.x * blockDim.x + threadIdx.x;
  const long long nThreads = (long long)gridDim.x * blockDim.x;
  const int r = threadIdx.x & 15;                 // lane within edge-group
  for (long long e = gid >> 4; e < nE; e += (nThreads >> 4)) {
    const int   s = src[e];
    const int   d = dst[e];
    const float v = vals[e];
    const float4 xv = *((const float4*)(x + (long long)s * DIM) + r);
    float* yp = y + (long long)d * DIM + r * 4;
    unsafeAtomicAdd(yp + 0, v * xv.x);
    unsafeAtomicAdd(yp + 1, v * xv.y);
    unsafeAtomicAdd(yp + 2, v * xv.z);
    unsafeAtomicAdd(yp + 3, v * xv.w);
  }
}

// ---------------------------------------------------------------------------
// LeakyReLU(0.5) in place, float4-vectorized.
// ---------------------------------------------------------------------------
__global__ __launch_bounds__(256) void leaky_kernel(float* __restrict__ x, long long n4)
{
  const long long i = (long long)blockIdx.x * blockDim.x + threadIdx.x;
  if (i < n4) {
    float4* p = (float4*)x + i;
    float4 v = *p;
    v.x = v.x > 0.f ? v.x : LEAKY_SLOPE * v.x;
    v.y = v.y > 0.f ? v.y : LEAKY_SLOPE * v.y;
    v.z = v.z > 0.f ? v.z : LEAKY_SLOPE * v.z;
    v.w = v.w > 0.f ? v.w : LEAKY_SLOPE * v.w;
    *p = v;
  }
}

// ---------------------------------------------------------------------------
// Final layer: out = LayerNorm(h) * gamma + beta + ego, per row of 64.
// One wave32 handles a 16-row tile. Row sums / sums-of-squares are computed
// with V_WMMA_F32_16X16X4_F32 against an all-ones B matrix:
//   A layout (16x4 f32): lanes 0-15 hold M=lane, K={0,1} in 2 VGPRs;
//                        lanes 16-31 hold M=lane-16, K={2,3}.
//   C layout (16x16 f32): lanes 0-15 vgpr v = C[v, lane]   (rows 0-7)
//                         lanes 16-31 vgpr v = C[8+v, ...] (rows 8-15)
// With B = ones, every column of C equals the row sum. EXEC stays all-ones
// through the WMMA region (guards are wave-uniform and only gate stores).
// ---------------------------------------------------------------------------
__global__ __launch_bounds__(256) void ln_residual_wmma_kernel(
    const float* __restrict__ h, const float* __restrict__ ego,
    const float* __restrict__ gamma, const float* __restrict__ beta,
    float* __restrict__ out, int nTiles)
{
  __shared__ float smem[8][32];                  // per wave: 16 sums + 16 sumsq
  const int wave = threadIdx.x >> 5;             // 0..7
  const int lane = threadIdx.x & 31;
  const int tile = blockIdx.x * 8 + wave;
  const bool active = tile < nTiles;
  const int t = active ? tile : 0;               // inactive waves shadow tile 0

  const int m    = lane & 15;                    // row within tile
  const int half = lane >> 4;                    // 0: K={0,1}, 1: K={2,3}
  const long long rowbase = ((long long)t * 16 + m) * DIM;

  v2f a[16];
  v8f csum = {};
  v8f csq  = {};
  v2f ones; ones[0] = 1.0f; ones[1] = 1.0f;

#pragma unroll
  for (int kc = 0; kc < 16; ++kc) {
    const int col = kc * 4 + half * 2;
    a[kc] = *(const v2f*)(h + rowbase + col);
    v2f a2; a2[0] = a[kc][0] * a[kc][0]; a2[1] = a[kc][1] * a[kc][1];
    csum = __builtin_amdgcn_wmma_f32_16x16x4_f32(false, a[kc], false, ones,
                                                 (short)0, csum, false, false);
    csq  = __builtin_amdgcn_wmma_f32_16x16x4_f32(false, a2,    false, ones,
                                                 (short)0, csq,  false, false);
  }

  // Lane 0 holds S[0..7] in its 8 C VGPRs, lane 16 holds S[8..15]; spill both.
  if (lane == 0 || lane == 16) {
#pragma unroll
    for (int v = 0; v < 8; ++v) {
      smem[wave][half * 8 + v]      = csum[v];
      smem[wave][16 + half * 8 + v] = csq[v];
    }
  }
  __syncthreads();

  const float S    = smem[wave][m];
  const float SQ   = smem[wave][16 + m];
  const float mu   = S * (1.0f / DIM);
  const float var  = SQ * (1.0f / DIM) - mu * mu;
  const float rstd = rsqrtf(var + LN_EPS);

  if (active) {
#pragma unroll
    for (int kc = 0; kc < 16; ++kc) {
      const int col = kc * 4 + half * 2;
      const v2f g = *(const v2f*)(gamma + col);
      const v2f b = *(const v2f*)(beta + col);
      const v2f rr = *(const v2f*)(ego + rowbase + col);
      v2f o;
      o[0] = (a[kc][0] - mu) * rstd * g[0] + b[0] + rr[0];
      o[1] = (a[kc][1] - mu) * rstd * g[1] + b[1] + rr[1];
      *(v2f*)(out + rowbase + col) = o;
    }
  }
}

// ---------------------------------------------------------------------------
// Host-side orchestration (all on `stream`, graph-capture safe).
// Workspace: two ping-pong N x D f32 buffers (76.8 MB).
// ---------------------------------------------------------------------------
extern "C" void kernel_launch(void* const* d_in, const int* in_sizes, int n_in,
                              void* d_out, int out_size, void* d_ws, size_t ws_size,
                              hipStream_t stream)
{
  const float* ego   = (const float*)d_in[0];
  const float* vals  = (const float*)d_in[1];
  const float* gamma = (const float*)d_in[2];
  const float* beta  = (const float*)d_in[3];
  const int*   rows  = (const int*)d_in[4];
  const int*   cols  = (const int*)d_in[5];
  const int    nE    = in_sizes[1];

  const long long ND = (long long)N_NODES * DIM;
  float* buf0 = (float*)d_ws;        // holds t = A^T e
  float* buf1 = buf0 + ND;           // holds h = A t (becomes next e)

  const int spmmBlocks = (int)(((long long)nE * 16 + 255) / 256);
  const long long n4 = ND / 4;
  const int elwBlocks = (int)((n4 + 255) / 256);
  const int nTiles   = N_NODES / 16;           // 9375, exact
  const int lnBlocks = (nTiles + 7) / 8;

  const float* e_in = ego;
  for (int layer = 0; layer < 3; ++layer) {
    // t = A^T e  : y[cols[e]] += vals[e] * x[rows[e]]
    hipMemsetAsync(buf0, 0, ND * sizeof(float), stream);
    spmm_scatter_kernel<<<spmmBlocks, 256, 0, stream>>>(cols, rows, vals, e_in, buf0, nE);
    // h = A t    : y[rows[e]] += vals[e] * x[cols[e]]
    hipMemsetAsync(buf1, 0, ND * sizeof(float), stream);
    spmm_scatter_kernel<<<spmmBlocks, 256, 0, stream>>>(rows, cols, vals, buf0, buf1, nE);
    if (layer < 2) {
      leaky_kernel<<<elwBlocks, 256, 0, stream>>>(buf1, n4);
      e_in = buf1;
    }
  }
  ln_residual_wmma_kernel<<<lnBlocks, 256, 0, stream>>>(
      buf1, ego, gamma, beta, (float*)d_out, nTiles);
}